// RCNN_12154757447922
// MI455X (gfx1250) — compile-verified
//
#include <hip/hip_runtime.h>
#include <hip/hip_bf16.h>
#include <math.h>

typedef __attribute__((ext_vector_type(2))) float v2f;
typedef __attribute__((ext_vector_type(8))) float v8f;
typedef int async_v2i __attribute__((vector_size(8)));

#define CC     512
#define HH     37
#define WW     50
#define PH     39           // padded H
#define PW     52           // padded W
#define HW     1850
#define BB     8
#define MTOT   14800        // B*H*W = 925*16 exactly
#define MTILES 925
#define K3     4608         // 512*9
#define NANCH  16650        // H*W*9
#define KTOP   400
#define ASTRIDE 514         // LDS row stride (floats) to avoid bank conflicts

// ---- CDNA5 async global->LDS copy (guarded; falls back to plain ds store) --
#if defined(__has_builtin)
#if __has_builtin(__builtin_amdgcn_global_load_async_to_lds_b64)
#define HAVE_ASYNC_LDS 1
#endif
#endif

#ifdef HAVE_ASYNC_LDS
// Signature (from clang diagnostic): (v2i AS1*, v2i AS3*, Imm offset, Imm cpol)
#define ASYNC_COPY_B64(g, l)                                              \
  __builtin_amdgcn_global_load_async_to_lds_b64(                          \
      (__attribute__((address_space(1))) async_v2i*)(g),                  \
      (__attribute__((address_space(3))) async_v2i*)(l), 0, 0)
#if __has_builtin(__builtin_amdgcn_s_wait_asynccnt)
#define ASYNC_WAIT() __builtin_amdgcn_s_wait_asynccnt(0)
#else
#define ASYNC_WAIT() asm volatile("s_wait_asynccnt 0x0" ::: "memory")
#endif
#else
#define ASYNC_COPY_B64(g, l) (*(float2*)(l) = *(const float2*)(g))
#define ASYNC_WAIT()
#endif

// ---------------------------------------------------------------------------
// Anchor geometry, recomputed on the fly. j = ratio_idx*3 + scale_idx.
// ---------------------------------------------------------------------------
__device__ __forceinline__ void anchor_box(int a, float& x1, float& y1,
                                           float& x2, float& y2) {
  int yy  = a / 450;
  int rem = a - yy * 450;
  int xx  = rem / 9;
  int j   = rem - xx * 9;
  int ri  = j / 3;
  int si  = j - ri * 3;
  float cx = xx * 16.0f + 8.0f;
  float cy = yy * 16.0f + 8.0f;
  float sr = (ri == 0) ? 0.70710678118654752f
                       : ((ri == 1) ? 1.0f : 1.41421356237309504f);
  float sc = (si == 0) ? 8.0f : ((si == 1) ? 16.0f : 24.0f);
  float h = 16.0f * sc * sr;
  float w = 16.0f * sc / sr;
  x1 = cx - w * 0.5f;
  y1 = cy - h * 0.5f;
  x2 = cx + w * 0.5f;
  y2 = cy + h * 0.5f;
}

// ---------------------------------------------------------------------------
// Zero the padded NHWC feature buffer (borders stay zero after packing).
// ---------------------------------------------------------------------------
__global__ __launch_bounds__(256) void zero_pad(float4* __restrict__ fp) {
  fp[blockIdx.x * 256 + threadIdx.x] = make_float4(0.f, 0.f, 0.f, 0.f);
}

// ---------------------------------------------------------------------------
// NCHW -> padded NHWC via LDS transpose. Block = (b, y, 64-channel tile).
// Reads coalesced over x, writes coalesced over c.
// ---------------------------------------------------------------------------
__global__ __launch_bounds__(256) void pack_feat(const float* __restrict__ feat,
                                                 float* __restrict__ fp) {
  __shared__ float tile[64][51];
  const int bid = blockIdx.x;            // 8*37*8
  const int ct  = bid & 7;
  const int rem = bid >> 3;
  const int y   = rem % HH;
  const int b   = rem / HH;
  const int t   = threadIdx.x;
#pragma unroll
  for (int i = 0; i < 13; ++i) {
    int idx = i * 256 + t;
    if (idx < 3200) {
      int cl = idx / WW;
      int x  = idx - cl * WW;
      tile[cl][x] = feat[((b * CC + ct * 64 + cl) * HH + y) * WW + x];
    }
  }
  __syncthreads();
#pragma unroll
  for (int i = 0; i < 13; ++i) {
    int idx = i * 256 + t;
    if (idx < 3200) {
      int x  = idx >> 6;
      int cl = idx & 63;
      fp[((b * PH + (y + 1)) * PW + (x + 1)) * CC + ct * 64 + cl] = tile[cl][x];
    }
  }
}

// ---------------------------------------------------------------------------
// Weight repack: dst[n][r][ci] = src[n][ci][r]  (r = ky*3+kx).
// Makes conv B-fragment pairs contiguous in memory.
// ---------------------------------------------------------------------------
__global__ __launch_bounds__(256) void pack_wgt(const float* __restrict__ src,
                                                float* __restrict__ dst) {
  int idx = blockIdx.x * 256 + threadIdx.x;   // 512*4608 = 2,359,296 exact
  int n   = idx / K3;
  int rem = idx - n * K3;
  int r   = rem >> 9;
  int ci  = rem & 511;
  dst[idx] = src[n * K3 + ci * 9 + r];
}

// ---------------------------------------------------------------------------
// 3x3 conv 512->512 as implicit GEMM on V_WMMA_F32_16X16X4_F32.
// K ordered as r*512+ci: per 3x3 tap the 16x512 A panel is async-copied to
// LDS once per block (shared by all 8 waves); no predicates, no div/mod in
// the hot loop. One wave owns a 16x64 tile (4 accumulators reuse one A frag).
// Output written NHWC [m][512].
// ---------------------------------------------------------------------------
__global__ __launch_bounds__(256) void conv3x3_wmma(
    const float* __restrict__ featp, const float* __restrict__ wgt,
    const float* __restrict__ bias, float* __restrict__ out)
{
  __shared__ float atile[16 * ASTRIDE];
  const int mtile = blockIdx.x;             // 925
  const int wv    = threadIdx.x >> 5;       // 8 waves: 64 channels each
  const int lane  = threadIdx.x & 31;
  const int mr    = lane & 15;
  const int khalf = (lane >> 4) << 1;       // 0 or 2 (A/B frag K-half)
  const int n0    = wv * 64 + mr;

  // staging role: 16 threads per row, b64 pairs over 512 channels
  const int srow = threadIdx.x >> 4;
  const int scol = (threadIdx.x & 15) << 1;
  const int sm   = mtile * 16 + srow;
  const int sb   = sm / HW;
  const int sp   = sm - sb * HW;
  const int sy   = sp / WW;
  const int sx   = sp - sy * WW;
  const float* sbase = featp + ((sb * PH + sy) * PW + sx) * CC + scol;
  float*       dbase = &atile[srow * ASTRIDE + scol];

  v8f acc[4] = {};
  const float* wbase = wgt + n0 * K3;
  for (int r = 0; r < 9; ++r) {
    const int ky = r / 3;
    const int kx = r - ky * 3;
    const float* srcr = sbase + (ky * PW + kx) * CC;   // padded: always valid
#pragma unroll
    for (int i = 0; i < 16; ++i)
      ASYNC_COPY_B64(srcr + i * 32, dbase + i * 32);
    ASYNC_WAIT();
    __syncthreads();

    const float* wr = wbase + r * 512 + khalf;
    const float* ar = &atile[mr * ASTRIDE + khalf];
#pragma unroll 4
    for (int c0 = 0; c0 < CC; c0 += 4) {
      v2f afrag = *(const v2f*)(ar + c0);
#pragma unroll
      for (int nt = 0; nt < 4; ++nt) {
        v2f bfrag = *(const v2f*)(wr + nt * 16 * K3 + c0);
        acc[nt] = __builtin_amdgcn_wmma_f32_16x16x4_f32(
            false, afrag, false, bfrag, (short)0, acc[nt], false, false);
      }
    }
    __syncthreads();
  }
  // D layout: VGPR rr -> M = rr (+8 for upper lane half), N = lane&15.
#pragma unroll
  for (int nt = 0; nt < 4; ++nt) {
    int n = wv * 64 + nt * 16 + (lane & 15);
    float bv = bias[n];
#pragma unroll
    for (int rr = 0; rr < 8; ++rr) {
      int mm = mtile * 16 + rr + ((lane >> 4) << 3);
      out[mm * CC + n] = acc[nt][rr] + bv;
    }
  }
}

// ---------------------------------------------------------------------------
// BatchNorm stats, two-stage, fully coalesced on NHWC (thread = channel).
// ---------------------------------------------------------------------------
#define BNB 116   // ceil(14800/128)
__global__ __launch_bounds__(512) void bn_partial(const float* __restrict__ convn,
                                                  float* __restrict__ part,
                                                  float* __restrict__ part2) {
  const int c  = threadIdx.x;
  const int g  = blockIdx.x;
  const int i0 = g * 128;
  const int i1 = (i0 + 128 < MTOT) ? i0 + 128 : MTOT;
  float s = 0.0f, s2 = 0.0f;
  for (int i = i0; i < i1; ++i) {
    float v = convn[i * CC + c];
    s  += v;
    s2 += v * v;
  }
  part[g * CC + c]  = s;
  part2[g * CC + c] = s2;
}

__global__ __launch_bounds__(512) void bn_finalize(
    const float* __restrict__ part, const float* __restrict__ part2,
    const float* __restrict__ gamma, const float* __restrict__ beta,
    float* __restrict__ scale, float* __restrict__ shift) {
  const int c = threadIdx.x;
  float s = 0.0f, s2 = 0.0f;
  for (int g = 0; g < BNB; ++g) {
    s  += part[g * CC + c];
    s2 += part2[g * CC + c];
  }
  float mean = s * (1.0f / MTOT);
  float var  = s2 * (1.0f / MTOT) - mean * mean;
  float sc   = gamma[c] * rsqrtf(var + 1e-5f);
  scale[c] = sc;
  shift[c] = beta[c] - mean * sc;
}

// ---------------------------------------------------------------------------
// 1x1 conv heads as WMMA GEMM; BN+ReLU fused into the A-fragment load
// (scale/shift staged in LDS). Epilogue writes anchor-major layout + bias.
// ---------------------------------------------------------------------------
__global__ __launch_bounds__(256) void head_wmma(
    const float* __restrict__ conv, const float* __restrict__ scale,
    const float* __restrict__ shift, const float* __restrict__ wgt,
    const float* __restrict__ bias, int N, int comps,
    float* __restrict__ outbase, int ntiles)
{
  __shared__ float s_sc[CC], s_sh[CC];
  for (int i = threadIdx.x; i < CC; i += 256) {
    s_sc[i] = scale[i];
    s_sh[i] = shift[i];
  }
  __syncthreads();

  const int wave = blockIdx.x * 8 + (threadIdx.x >> 5);
  if (wave >= MTILES * ntiles) return;      // wave-uniform: EXEC stays all-1s
  const int mtile = wave / ntiles;
  const int ntile = wave - mtile * ntiles;
  const int lane  = threadIdx.x & 31;
  const int mr    = lane & 15;
  const int m     = mtile * 16 + mr;
  const int khalf = (lane >> 4) << 1;
  const int ncol  = ntile * 16 + mr;

  v8f acc = {};
  const float* abase = conv + m * CC + khalf;   // NHWC: contiguous b64
  for (int k0 = 0; k0 < CC; k0 += 4) {
    v2f cv = *(const v2f*)(abase + k0);
    v2f afrag;
#pragma unroll
    for (int u = 0; u < 2; ++u) {
      int k = k0 + khalf + u;
      afrag[u] = fmaxf(cv[u] * s_sc[k] + s_sh[k], 0.0f);  // BN + ReLU fused
    }
    v2f bfrag = {};
    if (ncol < N) bfrag = *(const v2f*)(wgt + ncol * CC + k0 + khalf);
    acc = __builtin_amdgcn_wmma_f32_16x16x4_f32(
        false, afrag, false, bfrag, (short)0, acc, false, false);
  }
#pragma unroll
  for (int rr = 0; rr < 8; ++rr) {
    int n = ntile * 16 + (lane & 15);
    if (n < N) {
      int mm = mtile * 16 + rr + ((lane >> 4) << 3);
      int bb = mm / HW;
      int pp = mm - bb * HW;
      int j  = n / comps;
      int c  = n - j * comps;
      outbase[bb * (NANCH * comps) + (pp * 9 + j) * comps + c] = acc[rr] + bias[n];
    }
  }
}

// ---------------------------------------------------------------------------
// Top-400 per image. Invalid anchors forced to -1 (< any sigmoid) so top-k
// over all anchors == top-k over the valid subset. Register-resident chunks,
// LDS argmax tree, lower-index tie-break (matches jax.lax.top_k).
// ---------------------------------------------------------------------------
__global__ __launch_bounds__(1024) void topk_kernel(
    const float* __restrict__ cls, float* __restrict__ topv,
    int* __restrict__ topi)
{
  const int b = blockIdx.x;
  const int t = threadIdx.x;
  __shared__ float rv[1024];
  __shared__ int   ri[1024];
  __shared__ int   winner;
  const int PER = 17;
  const int beg = t * PER;

  float lv[PER];
#pragma unroll
  for (int u = 0; u < PER; ++u) {
    int a = beg + u;
    float s = -3.0f;
    if (a < NANCH) {
      float x1, y1, x2, y2;
      anchor_box(a, x1, y1, x2, y2);
      bool valid = (x1 >= 0.0f) && (y1 >= 0.0f) && (x2 < 800.0f) && (y2 < 592.0f);
      float c0 = cls[b * (NANCH * 2) + a * 2 + 0];
      float c1 = cls[b * (NANCH * 2) + a * 2 + 1];
      float sm = 1.0f / (1.0f + expf(c0 - c1));
      s = valid ? sm : -1.0f;
    }
    lv[u] = s;
  }
  float bv = -4.0f; int bi = NANCH;
#pragma unroll
  for (int u = 0; u < PER; ++u)
    if (lv[u] > bv) { bv = lv[u]; bi = beg + u; }

  for (int sel = 0; sel < KTOP; ++sel) {
    rv[t] = bv; ri[t] = bi;
    __syncthreads();
    for (int s = 512; s > 0; s >>= 1) {
      if (t < s) {
        float ov = rv[t + s]; int oi = ri[t + s];
        if (ov > rv[t] || (ov == rv[t] && oi < ri[t])) { rv[t] = ov; ri[t] = oi; }
      }
      __syncthreads();
    }
    if (t == 0) {
      winner = ri[0];
      topv[b * KTOP + sel] = rv[0];
      topi[b * KTOP + sel] = ri[0];
    }
    __syncthreads();
    int wnr = winner;
    if (wnr >= beg && wnr < beg + PER) {
#pragma unroll
      for (int u = 0; u < PER; ++u)
        if (beg + u == wnr) lv[u] = -5.0f;
      bv = -4.0f; bi = NANCH;
#pragma unroll
      for (int u = 0; u < PER; ++u)
        if (lv[u] > bv) { bv = lv[u]; bi = beg + u; }
    }
    __syncthreads();
  }
}

// ---------------------------------------------------------------------------
// Box decode + greedy NMS (scores pre-sorted -> argsort is identity).
// ---------------------------------------------------------------------------
__global__ __launch_bounds__(512) void decode_nms(
    const float* __restrict__ offp, const int* __restrict__ topi,
    float* __restrict__ boxes, float* __restrict__ keep)
{
  const int b = blockIdx.x;
  const int t = threadIdx.x;
  __shared__ float tx1[KTOP], ty1[KTOP], tx2[KTOP], ty2[KTOP], tar[KTOP];
  __shared__ int   supp[KTOP];
  if (t < KTOP) {
    int a = topi[b * KTOP + t];
    float a0, a1, a2, a3;
    anchor_box(a, a0, a1, a2, a3);
    float xa = (a0 + a2) * 0.5f;
    float ya = (a1 + a3) * 0.5f;
    float wa = a2 - a0 + 1.0f;
    float ha = a3 - a1 + 1.0f;
    const float* tp = offp + b * (NANCH * 4) + a * 4;
    float t0 = tp[0], t1 = tp[1], t2 = tp[2], t3 = tp[3];
    float xc = t0 * wa + xa;
    float yc = t1 * ha + ya;
    float wb = wa * expf(t2);
    float hb = ha * expf(t3);
    float X1 = xc - wb * 0.5f, Y1 = yc - hb * 0.5f;
    float X2 = xc + wb * 0.5f, Y2 = yc + hb * 0.5f;
    float* bo = boxes + b * (KTOP * 4) + t * 4;
    bo[0] = X1; bo[1] = Y1; bo[2] = X2; bo[3] = Y2;
    float u1 = truncf(X1), v1 = truncf(Y1), u2 = truncf(X2), v2 = truncf(Y2);
    tx1[t] = u1; ty1[t] = v1; tx2[t] = u2; ty2[t] = v2;
    tar[t] = (u2 - u1 + 1.0f) * (v2 - v1 + 1.0f);
    supp[t] = 0;
  }
  __syncthreads();
  for (int i = 0; i < KTOP; ++i) {
    bool alive = (supp[i] == 0);
    if (alive && t < KTOP && t > i) {
      float iw = fminf(tx2[i], tx2[t]) - fmaxf(tx1[i], tx1[t]) + 1.0f;
      float ih = fminf(ty2[i], ty2[t]) - fmaxf(ty1[i], ty1[t]) + 1.0f;
      iw = fmaxf(iw, 0.0f);
      ih = fmaxf(ih, 0.0f);
      float inter = iw * ih;
      float iou = inter / (tar[i] + tar[t] - inter);
      if (iou > 0.6f) supp[t] = 1;
    }
    __syncthreads();
  }
  if (t < KTOP) keep[b * KTOP + t] = supp[t] ? 0.0f : 1.0f;
}

// ---------------------------------------------------------------------------
extern "C" void kernel_launch(void* const* d_in, const int* in_sizes, int n_in,
                              void* d_out, int out_size, void* d_ws, size_t ws_size,
                              hipStream_t stream) {
  (void)in_sizes; (void)n_in; (void)out_size; (void)ws_size;
  const float* feature = (const float*)d_in[0];
  const float* md_w    = (const float*)d_in[1];
  const float* md_b    = (const float*)d_in[2];
  const float* gamma   = (const float*)d_in[3];
  const float* beta    = (const float*)d_in[4];
  const float* cls_w   = (const float*)d_in[5];
  const float* cls_b   = (const float*)d_in[6];
  const float* off_w   = (const float*)d_in[7];
  const float* off_b   = (const float*)d_in[8];

  // d_out layout (floats): offset_pred | cls_pred | boxes | keep
  float* out       = (float*)d_out;
  float* out_off   = out;                 // 8*16650*4 = 532800
  float* out_cls   = out + 532800;        // 8*16650*2 = 266400
  float* out_boxes = out + 799200;        // 8*400*4   = 12800
  float* out_keep  = out + 812000;        // 8*400     = 3200

  // workspace layout (floats)
  float* ws    = (float*)d_ws;
  float* convn = ws;                      // NHWC conv out: 14800*512 = 7,577,600
  float* featp = ws + 7577600;            // padded NHWC feat: 8*39*52*512 = 8,306,688
  float* wgt_t = ws + 15884288;           // packed 3x3 weights: 512*4608 = 2,359,296
  float* part  = ws + 18243584;           // BN partial sums: 116*512
  float* part2 = ws + 18302976;           // BN partial sumsq: 116*512
  float* scale = ws + 18362368;           // 512
  float* shift = ws + 18362880;           // 512
  float* topv  = ws + 18363392;           // 8*400
  int*   topi  = (int*)(ws + 18366592);   // 8*400

  zero_pad<<<8112, 256, 0, stream>>>((float4*)featp);          // 8,306,688 floats
  pack_feat<<<BB * HH * 8, 256, 0, stream>>>(feature, featp);  // 2368 blocks
  pack_wgt<<<9216, 256, 0, stream>>>(md_w, wgt_t);             // 2,359,296 exact
  conv3x3_wmma<<<MTILES, 256, 0, stream>>>(featp, wgt_t, md_b, convn);
  bn_partial<<<BNB, 512, 0, stream>>>(convn, part, part2);
  bn_finalize<<<1, 512, 0, stream>>>(part, part2, gamma, beta, scale, shift);
  head_wmma<<<(MTILES * 3 + 7) / 8, 256, 0, stream>>>(convn, scale, shift,
      off_w, off_b, 36, 4, out_off, 3);
  head_wmma<<<(MTILES * 2 + 7) / 8, 256, 0, stream>>>(convn, scale, shift,
      cls_w, cls_b, 18, 2, out_cls, 2);
  topk_kernel<<<BB, 1024, 0, stream>>>(out_cls, topv, topi);
  decode_nms<<<BB, 512, 0, stream>>>(out_off, topi, out_boxes, out_keep);
}